// GATAuto_30940944401095
// MI455X (gfx1250) — compile-verified
//
#include <hip/hip_runtime.h>
#include <stdint.h>

typedef __attribute__((ext_vector_type(2))) float v2f;
typedef __attribute__((ext_vector_type(8))) float v8f;

#define NEGV (-9.0e15f)
#define NN 2048
#define BB 8

__device__ __forceinline__ v8f wmma_f32(v2f a, v2f b, v8f c) {
  return __builtin_amdgcn_wmma_f32_16x16x4_f32(false, a, false, b, (short)0, c,
                                               false, false);
}

// ---------------------------------------------------------------- mask pack
// adj[N,N] -> bitmask words mask[i*64 + j/32], bit j%32 = (adj[i][j] > 0)
__global__ void mask_kernel(const float* __restrict__ adj,
                            uint32_t* __restrict__ mask) {
  int lane = threadIdx.x & 31;
  int gwave = (blockIdx.x * blockDim.x + threadIdx.x) >> 5;
  for (int w = gwave; w < NN * (NN / 32); w += 8192) {
    float v = adj[(size_t)w * 32 + lane];
    unsigned long long bal = __ballot(v > 0.0f);
    if (lane == 0) mask[w] = (uint32_t)bal;
  }
}

// ---------------------------------------------------------------- h = X @ W
// X:[B*N,128] row-major, W:[128,F], H:[B*N,F].  grid = B*(N/16),
// block = 2*F threads (F/16 waves, one 16-col tile per wave).
template <int FT>
__global__ void gemm_h(const float* __restrict__ X, const float* __restrict__ W,
                       float* __restrict__ H) {
  const int Fi = 128;
  int blk = blockIdx.x;
  int b = blk >> 7;                 // N/16 == 128
  int ibase = (blk & 127) * 16;
  int tid = threadIdx.x;
  int wave = tid >> 5, lane = tid & 31;
  __shared__ __align__(16) float Xs[16 * 130];

  const float* Xrow = X + ((size_t)b * NN + ibase) * Fi;
  for (int idx = tid; idx < 16 * 128; idx += 2 * FT) {
    int r = idx >> 7, c = idx & 127;
    Xs[r * 130 + c] = Xrow[(size_t)r * Fi + c];
  }
  __syncthreads();

  int M = lane & 15;
  int klo = (lane < 16) ? 0 : 2;
  int colg = wave * 16 + M;
  v8f acc = {};
  const float2* Xs2 = (const float2*)Xs;
  for (int kb = 0; kb < Fi; kb += 4) {
    int ka = kb + klo;
    float2 af = Xs2[(M * 130 + ka) >> 1];       // 8B aligned (even index)
    v2f a; a.x = af.x; a.y = af.y;
    v2f bb;
    bb.x = W[(size_t)ka * FT + colg];
    bb.y = W[(size_t)(ka + 1) * FT + colg];
    acc = wmma_f32(a, bb, acc);
  }

  float* Hrow = H + ((size_t)b * NN + ibase) * FT;
  int rofs = (lane < 16) ? 0 : 8;
#pragma unroll
  for (int r = 0; r < 8; r++)
    Hrow[(size_t)(r + rofs) * FT + wave * 16 + M] = acc[r];
}

// ------------------------------------------------- es = h@a_src, ed = h@a_dst
// one wave per row; rows = B*N.  grid = 2048, block = 256.
template <int FT>
__global__ void esed_kernel(const float* __restrict__ H,
                            const float* __restrict__ as,
                            const float* __restrict__ ad,
                            float* __restrict__ es, float* __restrict__ ed) {
  int tid = threadIdx.x, wave = tid >> 5, lane = tid & 31;
  int row = blockIdx.x * 8 + wave;
  const float* Hr = H + (size_t)row * FT;
  float ssum = 0.f, dsum = 0.f;
  for (int t = lane; t < FT; t += 32) {
    float hv = Hr[t];
    ssum += hv * as[t];
    dsum += hv * ad[t];
  }
  for (int off = 16; off; off >>= 1) {
    ssum += __shfl_xor(ssum, off, 32);
    dsum += __shfl_xor(dsum, off, 32);
  }
  if (lane == 0) { es[row] = ssum; ed[row] = dsum; }
}

// ----------------------------------------------- row max + 1/sum for softmax
// grid = B*(N/8) = 2048 blocks, block = 256 (8 waves, one row each)
__global__ void stats_kernel(const float* __restrict__ es,
                             const float* __restrict__ ed,
                             const uint32_t* __restrict__ mask,
                             float* __restrict__ mrow,
                             float* __restrict__ rsrow) {
  int b = blockIdx.x >> 8;
  int rowbase = (blockIdx.x & 255) * 8;
  int tid = threadIdx.x, wave = tid >> 5, lane = tid & 31;
  __shared__ __align__(16) float eds[NN];
  const float* edb = ed + b * NN;
  for (int idx = tid; idx < NN; idx += 256) eds[idx] = edb[idx];
  __syncthreads();

  int i = rowbase + wave;
  float es_i = es[b * NN + i];
  const uint32_t* mp = mask + (size_t)i * 64;

  float mx = NEGV;
  for (int jw = 0; jw < 64; jw++) {
    uint32_t w = mp[jw];
    float e = es_i + eds[jw * 32 + lane];
    e = (e > 0.f) ? e : 0.2f * e;
    float v = ((w >> lane) & 1u) ? e : NEGV;
    mx = fmaxf(mx, v);
  }
  for (int off = 16; off; off >>= 1) mx = fmaxf(mx, __shfl_xor(mx, off, 32));

  float s = 0.f;
  for (int jw = 0; jw < 64; jw++) {
    uint32_t w = mp[jw];
    float e = es_i + eds[jw * 32 + lane];
    e = (e > 0.f) ? e : 0.2f * e;
    float v = ((w >> lane) & 1u) ? e : NEGV;
    s += __expf(v - mx);
  }
  for (int off = 16; off; off >>= 1) s += __shfl_xor(s, off, 32);

  if (lane == 0) { mrow[b * NN + i] = mx; rsrow[b * NN + i] = 1.0f / s; }
}

// ------------------------------------------- out = relu(softmax(att) @ h)
// grid = B*32 blocks (64 rows each), block = 128 (4 waves x 16 rows)
template <int FT>
__global__ void att_kernel(const float* __restrict__ H,
                           const float* __restrict__ es,
                           const float* __restrict__ ed,
                           const float* __restrict__ mrow,
                           const float* __restrict__ rsrow,
                           const uint32_t* __restrict__ mask,
                           float* __restrict__ O) {
  constexpr int NT = FT / 16;
  constexpr int LOGF = (FT == 128) ? 7 : 6;
  int b = blockIdx.x >> 5;
  int ibase = (blockIdx.x & 31) * 64;
  int tid = threadIdx.x, wave = tid >> 5, lane = tid & 31;

  __shared__ __align__(16) float eds[NN];
  __shared__ __align__(16) float hTs[FT * 18];   // transposed chunk, pad 18

  const float* edb = ed + b * NN;
  for (int idx = tid; idx < NN; idx += 128) eds[idx] = edb[idx];

  int M = lane & 15;
  int row = ibase + wave * 16 + M;
  float es_r = es[b * NN + row];
  float m_r  = mrow[b * NN + row];
  float rs_r = rsrow[b * NN + row];
  const uint32_t* mrowp = mask + (size_t)row * 64;
  int klo = (lane < 16) ? 0 : 2;

  v8f acc[NT];
#pragma unroll
  for (int ct = 0; ct < NT; ct++) acc[ct] = (v8f){};

  const float* Hb = H + (size_t)b * NN * FT;
  const float2* eds2 = (const float2*)eds;
  const float2* hTs2 = (const float2*)hTs;

  for (int jc = 0; jc < NN / 16; jc++) {
    int jbase = jc << 4;
    __syncthreads();
    for (int idx = tid; idx < FT * 16; idx += 128) {
      int col = idx & (FT - 1);
      int k = idx >> LOGF;
      hTs[col * 18 + k] = Hb[(size_t)(jbase + k) * FT + col];
    }
    __syncthreads();

    uint32_t word = mrowp[jbase >> 5];
    uint32_t bits = (word >> (jbase & 31)) & 0xFFFFu;

#pragma unroll
    for (int t = 0; t < 4; t++) {
      int jk = 4 * t + klo;
      float2 edv = eds2[(jbase + jk) >> 1];
      float e0 = es_r + edv.x; e0 = (e0 > 0.f) ? e0 : 0.2f * e0;
      float e1 = es_r + edv.y; e1 = (e1 > 0.f) ? e1 : 0.2f * e1;
      float v0 = ((bits >> jk) & 1u) ? e0 : NEGV;
      float v1 = ((bits >> (jk + 1)) & 1u) ? e1 : NEGV;
      v2f a;
      a.x = __expf(v0 - m_r) * rs_r;
      a.y = __expf(v1 - m_r) * rs_r;
#pragma unroll
      for (int ct = 0; ct < NT; ct++) {
        int col = ct * 16 + M;
        float2 bv = hTs2[(col * 18 + jk) >> 1];   // even -> 8B aligned
        v2f bb; bb.x = bv.x; bb.y = bv.y;
        acc[ct] = wmma_f32(a, bb, acc[ct]);
      }
    }
  }

  float* Ob = O + ((size_t)b * NN + ibase + wave * 16) * FT;
  int rofs = (lane < 16) ? 0 : 8;
#pragma unroll
  for (int ct = 0; ct < NT; ct++)
#pragma unroll
    for (int r = 0; r < 8; r++) {
      float v = acc[ct][r];
      v = (v > 0.f) ? v : 0.f;
      Ob[(size_t)(r + rofs) * FT + ct * 16 + M] = v;
    }
}

// -------------------------------------------- final linear, split-K partials
// X3:[8,131072], Wl:[131072,128].  grid = 128 K-chunks, block = 128 (col).
__global__ void lin_partial(const float* __restrict__ X3,
                            const float* __restrict__ Wl,
                            float* __restrict__ part) {
  const int K = NN * 64;            // 131072
  int g = blockIdx.x;
  int col = threadIdx.x;
  int k0 = g * 1024;
  __shared__ __align__(16) float xs[8][1024];
  for (int idx = threadIdx.x; idx < 8 * 1024; idx += 128) {
    int m = idx >> 10, k = idx & 1023;
    xs[m][k] = X3[(size_t)m * K + k0 + k];
  }
  __syncthreads();
  float acc[8] = {0.f, 0.f, 0.f, 0.f, 0.f, 0.f, 0.f, 0.f};
  for (int k = 0; k < 1024; k++) {
    float wv = Wl[(size_t)(k0 + k) * 128 + col];
#pragma unroll
    for (int m = 0; m < 8; m++) acc[m] += xs[m][k] * wv;
  }
#pragma unroll
  for (int m = 0; m < 8; m++)
    part[((size_t)g * 8 + m) * 128 + col] = acc[m];
}

__global__ void lin_reduce(const float* __restrict__ part,
                           const float* __restrict__ blin,
                           float* __restrict__ out) {
  int idx = blockIdx.x * 256 + threadIdx.x;   // 0..1023 -> (m,col)
  int col = idx & 127;
  float s = blin[col];
  for (int g = 0; g < 128; g++) s += part[(size_t)g * 1024 + idx];
  out[idx] = s;
}

// ---------------------------------------------------------------- launcher
extern "C" void kernel_launch(void* const* d_in, const int* in_sizes, int n_in,
                              void* d_out, int out_size, void* d_ws,
                              size_t ws_size, hipStream_t stream) {
  const float* x    = (const float*)d_in[0];
  const float* adj  = (const float*)d_in[1];
  const float* W1   = (const float*)d_in[2];
  const float* a1s  = (const float*)d_in[3];
  const float* a1d  = (const float*)d_in[4];
  const float* W2   = (const float*)d_in[5];
  const float* a2s  = (const float*)d_in[6];
  const float* a2d  = (const float*)d_in[7];
  const float* W3   = (const float*)d_in[8];
  const float* a3s  = (const float*)d_in[9];
  const float* a3d  = (const float*)d_in[10];
  const float* Wlin = (const float*)d_in[11];
  const float* blin = (const float*)d_in[12];
  float* out = (float*)d_out;

  // workspace carve-up
  char* w = (char*)d_ws;
  uint32_t* mask = (uint32_t*)w;          w += (size_t)NN * 64 * 4;       // 512K
  float* hA = (float*)w;                  w += (size_t)BB * NN * 128 * 4; // 8M
  float* hB = (float*)w;                  w += (size_t)BB * NN * 128 * 4; // 8M
  float* es = (float*)w;                  w += (size_t)BB * NN * 4;
  float* ed = (float*)w;                  w += (size_t)BB * NN * 4;
  float* mr = (float*)w;                  w += (size_t)BB * NN * 4;
  float* rs = (float*)w;                  w += (size_t)BB * NN * 4;
  float* part = (float*)w;                w += (size_t)128 * 8 * 128 * 4; // 512K

  mask_kernel<<<1024, 256, 0, stream>>>(adj, mask);

  // ---- layer 1 (128 -> 128)
  gemm_h<128><<<BB * 128, 256, 0, stream>>>(x, W1, hA);
  esed_kernel<128><<<2048, 256, 0, stream>>>(hA, a1s, a1d, es, ed);
  stats_kernel<<<2048, 256, 0, stream>>>(es, ed, mask, mr, rs);
  att_kernel<128><<<BB * 32, 128, 0, stream>>>(hA, es, ed, mr, rs, mask, hB);

  // ---- layer 2 (128 -> 128)
  gemm_h<128><<<BB * 128, 256, 0, stream>>>(hB, W2, hA);
  esed_kernel<128><<<2048, 256, 0, stream>>>(hA, a2s, a2d, es, ed);
  stats_kernel<<<2048, 256, 0, stream>>>(es, ed, mask, mr, rs);
  att_kernel<128><<<BB * 32, 128, 0, stream>>>(hA, es, ed, mr, rs, mask, hB);

  // ---- layer 3 (128 -> 64)
  gemm_h<64><<<BB * 128, 128, 0, stream>>>(hB, W3, hA);
  esed_kernel<64><<<2048, 256, 0, stream>>>(hA, a3s, a3d, es, ed);
  stats_kernel<<<2048, 256, 0, stream>>>(es, ed, mask, mr, rs);
  att_kernel<64><<<BB * 32, 128, 0, stream>>>(hA, es, ed, mr, rs, mask, hB);

  // ---- final linear: [8,131072] @ [131072,128] + blin
  lin_partial<<<128, 128, 0, stream>>>(hB, Wlin, part);
  lin_reduce<<<4, 256, 0, stream>>>(part, blin, out);
}